// PTBlock_47072841564819
// MI455X (gfx1250) — compile-verified
//
#include <hip/hip_runtime.h>
#include <hip/hip_bf16.h>
#include <math.h>

typedef float v2f __attribute__((ext_vector_type(2)));
typedef float v8f __attribute__((ext_vector_type(8)));

#define B_ 4
#define N_ 4096
#define KNN 16
#define HCH 128
#define IND 64
#define BN_ (B_ * N_)          // 16384
#define EPS_ 1e-5f

// ---- CDNA5 async global->LDS copy (ASYNCcnt path, no VGPR round-trip) ----
__device__ __forceinline__ void async_ld_b128(unsigned lds_off, const void* g) {
  asm volatile("global_load_async_to_lds_b128 %0, %1, off"
               :: "v"(lds_off), "v"(g) : "memory");
}
__device__ __forceinline__ void wait_async0() {
  asm volatile("s_wait_asynccnt 0" ::: "memory");
}

// -------------------------------------------------------------------------
// 1) Ball query: per point, first 16 indices (ascending) with d2 <= R^2.
//    Coordinates are small integers -> fp32 arithmetic is exact, matching
//    the reference's sq[i]+sq[j]-2*dot formulation bit-exactly in compares.
// -------------------------------------------------------------------------
__global__ void __launch_bounds__(256) k_ballquery(const int* __restrict__ coords,
                                                   int* __restrict__ gidx) {
  extern __shared__ float lds[];
  float* sx = lds;
  float* sy = lds + N_;
  float* sz = lds + 2 * N_;
  const int blocksPerBatch = N_ / 256;
  int b = blockIdx.x / blocksPerBatch;
  int i0 = (blockIdx.x % blocksPerBatch) * 256;
  for (int j = threadIdx.x; j < N_; j += 256) {
    int base = (b * N_ + j) * 4;
    sx[j] = (float)coords[base + 1];
    sy[j] = (float)coords[base + 2];
    sz[j] = (float)coords[base + 3];
  }
  __syncthreads();
  int il = i0 + threadIdx.x;
  float px = sx[il], py = sy[il], pz = sz[il];
  int gi = b * N_ + il;
  int cnt = 0;
  int first = gi;
  for (int j = 0; j < N_; ++j) {
    float dx = px - sx[j], dy = py - sy[j], dz = pz - sz[j];
    float d2 = dx * dx + dy * dy + dz * dz;
    if (d2 <= 100.0f) {
      if (cnt == 0) first = b * N_ + j;
      gidx[gi * KNN + cnt] = b * N_ + j;
      cnt++;
      if (cnt >= KNN) break;
    }
  }
  for (int t = cnt; t < KNN; ++t) gidx[gi * KNN + t] = first;
}

// -------------------------------------------------------------------------
// 2) Generic fp32 WMMA GEMM:  C[M x N] = (A*scale+shift) @ Bw + bias
//    One wave per 16-row tile; NT = N/16 column tiles; K-step = 4.
//    Fragment layouts per CDNA5 ISA 7.12.2 (32-bit A 16x4, B 4x16, C/D 16x16).
//    Weights staged into LDS with GLOBAL_LOAD_ASYNC_TO_LDS_B128.
// -------------------------------------------------------------------------
template <int NT>
__global__ void __launch_bounds__(256) k_gemm(const float* __restrict__ A,
                                              const float* __restrict__ scale,
                                              const float* __restrict__ shift,
                                              const float* __restrict__ Bw,
                                              const float* __restrict__ bias,
                                              float* __restrict__ C, int M, int K) {
  const int N = NT * 16;
  extern __shared__ float sB[];
  {
    unsigned lbase = (unsigned)(size_t)sB;
    for (int t = threadIdx.x * 4; t < K * N; t += 1024)
      async_ld_b128(lbase + t * 4, Bw + t);
    wait_async0();
  }
  __syncthreads();

  int lane = threadIdx.x & 31;
  int w = threadIdx.x >> 5;
  int m16 = lane & 15;
  int h = lane >> 4;
  int row0 = blockIdx.x * 128 + w * 16;
  if (row0 >= M) return;

  v8f acc[NT] = {};
  const float* ap = A + (size_t)(row0 + m16) * K;
  for (int s = 0; s < K / 4; ++s) {
    int k0 = 4 * s + 2 * h;
    v2f a;
    a.x = ap[k0];
    a.y = ap[k0 + 1];
    if (scale) {
      a.x = a.x * scale[k0] + shift[k0];
      a.y = a.y * scale[k0 + 1] + shift[k0 + 1];
    }
#pragma unroll
    for (int tn = 0; tn < NT; ++tn) {
      v2f b;
      b.x = sB[k0 * N + tn * 16 + m16];
      b.y = sB[(k0 + 1) * N + tn * 16 + m16];
      acc[tn] = __builtin_amdgcn_wmma_f32_16x16x4_f32(false, a, false, b, (short)0,
                                                      acc[tn], false, false);
    }
  }
#pragma unroll
  for (int tn = 0; tn < NT; ++tn) {
    int c = tn * 16 + m16;
    float bi = bias ? bias[c] : 0.0f;
#pragma unroll
    for (int e = 0; e < 8; ++e) {
      C[(size_t)(row0 + e + 8 * h) * N + c] = acc[tn][e] + bi;
    }
  }
}

// -------------------------------------------------------------------------
// 3) Batchnorm statistics over axis 0 -> per-channel scale/shift.
// -------------------------------------------------------------------------
__global__ void __launch_bounds__(256) k_bnstats(const float* __restrict__ T,
                                                 const float* __restrict__ g,
                                                 const float* __restrict__ be,
                                                 float* __restrict__ scale,
                                                 float* __restrict__ shift, int M, int C) {
  __shared__ float s1[256];
  __shared__ float s2[256];
  int c = blockIdx.x;
  float sum = 0.f, sq = 0.f;
  for (int r = threadIdx.x; r < M; r += 256) {
    float v = T[(size_t)r * C + c];
    sum += v;
    sq += v * v;
  }
  s1[threadIdx.x] = sum;
  s2[threadIdx.x] = sq;
  __syncthreads();
  for (int off = 128; off > 0; off >>= 1) {
    if (threadIdx.x < off) {
      s1[threadIdx.x] += s1[threadIdx.x + off];
      s2[threadIdx.x] += s2[threadIdx.x + off];
    }
    __syncthreads();
  }
  if (threadIdx.x == 0) {
    float mean = s1[0] / (float)M;
    float var = s2[0] / (float)M - mean * mean;
    float sc = g[c] * rsqrtf(var + EPS_);
    scale[c] = sc;
    shift[c] = be[c] - mean * sc;
  }
}

// -------------------------------------------------------------------------
// 4) Fused per-point core: one wave per point (its 16 neighbors = one 16-row
//    WMMA tile). delta GEMM (Wd2), attention GEMMs (Wg1, Wg2), per-channel
//    softmax over neighbors, weighted aggregation. All three 128x128 weights
//    async-staged into LDS (192KB) + 8 per-wave 16x132 transpose buffers.
//    LDS float offsets: sWd2=0, sWg1=16384, sWg2=32768, sWd1=49152,
//    sbd1=49536, sbd2=49664, sbg1=49792, sbg2=49920, sabuf=50048, sg=66944.
// -------------------------------------------------------------------------
__global__ void __launch_bounds__(256) k_pt_core(
    const int* __restrict__ coords, const int* __restrict__ gidx,
    const float* __restrict__ q, const float* __restrict__ kf,
    const float* __restrict__ vmat, const float* __restrict__ Wd1,
    const float* __restrict__ bd1, const float* __restrict__ Wd2,
    const float* __restrict__ bd2, const float* __restrict__ Wg1,
    const float* __restrict__ bg1, const float* __restrict__ Wg2,
    const float* __restrict__ bg2, float* __restrict__ agg) {
  extern __shared__ float lds[];
  float* sWd2 = lds;                // 16384
  float* sWg1 = sWd2 + 16384;       // 16384
  float* sWg2 = sWg1 + 16384;       // 16384
  float* sWd1 = sWg2 + 16384;       // 384
  float* sbd1 = sWd1 + 384;         // 128
  float* sbd2 = sbd1 + 128;         // 128
  float* sbg1 = sbd2 + 128;         // 128
  float* sbg2 = sbg1 + 128;         // 128
  float* sabuf = sbg2 + 128;        // 8 waves * 16*132
  int* sg = (int*)(sabuf + 8 * 16 * 132);  // 8 waves * 16 neighbor ids

  {
    unsigned lbase = (unsigned)(size_t)lds;
    for (int t = threadIdx.x * 4; t < 16384; t += 1024) {
      async_ld_b128(lbase + t * 4, Wd2 + t);
      async_ld_b128(lbase + (16384 + t) * 4, Wg1 + t);
      async_ld_b128(lbase + (32768 + t) * 4, Wg2 + t);
    }
    for (int t = threadIdx.x * 4; t < 384; t += 1024)
      async_ld_b128(lbase + (49152 + t) * 4, Wd1 + t);
    if (threadIdx.x < 32) {
      int t = threadIdx.x * 4;
      async_ld_b128(lbase + (49536 + t) * 4, bd1 + t);
      async_ld_b128(lbase + (49664 + t) * 4, bd2 + t);
      async_ld_b128(lbase + (49792 + t) * 4, bg1 + t);
      async_ld_b128(lbase + (49920 + t) * 4, bg2 + t);
    }
    wait_async0();
  }
  int lane = threadIdx.x & 31;
  int w = threadIdx.x >> 5;
  int m16 = lane & 15;
  int h = lane >> 4;
  int i = blockIdx.x * 8 + w;
  if (lane < 16) sg[w * 16 + lane] = gidx[i * KNN + lane];
  __syncthreads();

  float* abuf = sabuf + w * (16 * 132);
  int gm = sg[w * 16 + m16];
  // relative position for neighbor row m16
  float r0 = (float)(coords[i * 4 + 1] - coords[gm * 4 + 1]);
  float r1 = (float)(coords[i * 4 + 2] - coords[gm * 4 + 2]);
  float r2 = (float)(coords[i * 4 + 3] - coords[gm * 4 + 3]);

  // ---- delta = relu(rel @ Wd1 + bd1) @ Wd2 + bd2   (kept in D-layout regs)
  v8f dacc[8] = {};
  for (int s = 0; s < 32; ++s) {
    int k0 = 4 * s + 2 * h;
    float p0 = r0 * sWd1[k0] + r1 * sWd1[128 + k0] + r2 * sWd1[256 + k0] + sbd1[k0];
    float p1 = r0 * sWd1[k0 + 1] + r1 * sWd1[128 + k0 + 1] + r2 * sWd1[256 + k0 + 1] +
               sbd1[k0 + 1];
    v2f a;
    a.x = fmaxf(p0, 0.f);
    a.y = fmaxf(p1, 0.f);
#pragma unroll
    for (int tn = 0; tn < 8; ++tn) {
      v2f b;
      b.x = sWd2[k0 * 128 + tn * 16 + m16];
      b.y = sWd2[(k0 + 1) * 128 + tn * 16 + m16];
      dacc[tn] = __builtin_amdgcn_wmma_f32_16x16x4_f32(false, a, false, b, (short)0,
                                                       dacc[tn], false, false);
    }
  }
#pragma unroll
  for (int tn = 0; tn < 8; ++tn) {
    float bi = sbd2[tn * 16 + m16];
#pragma unroll
    for (int e = 0; e < 8; ++e) dacc[tn][e] += bi;
  }

  // ---- a0 = q[i] - kf[g_m] + delta, assembled row-major in padded LDS tile
#pragma unroll
  for (int tn = 0; tn < 8; ++tn)
#pragma unroll
    for (int e = 0; e < 8; ++e)
      abuf[(e + 8 * h) * 132 + tn * 16 + m16] = dacc[tn][e];
  {
    const float* qp = q + (size_t)i * HCH;
    const float* kp = kf + (size_t)gm * HCH;
    for (int s = 0; s < 32; ++s) {
      int k0 = 4 * s + 2 * h;
      float2 q2 = *(const float2*)(qp + k0);
      float2 k2 = *(const float2*)(kp + k0);
      abuf[m16 * 132 + k0] += q2.x - k2.x;
      abuf[m16 * 132 + k0 + 1] += q2.y - k2.y;
    }
  }

  // ---- hgem = relu(a0 @ Wg1 + bg1)
  v8f acc[8] = {};
  for (int s = 0; s < 32; ++s) {
    int k0 = 4 * s + 2 * h;
    v2f a;
    a.x = abuf[m16 * 132 + k0];
    a.y = abuf[m16 * 132 + k0 + 1];
#pragma unroll
    for (int tn = 0; tn < 8; ++tn) {
      v2f b;
      b.x = sWg1[k0 * 128 + tn * 16 + m16];
      b.y = sWg1[(k0 + 1) * 128 + tn * 16 + m16];
      acc[tn] = __builtin_amdgcn_wmma_f32_16x16x4_f32(false, a, false, b, (short)0,
                                                      acc[tn], false, false);
    }
  }
#pragma unroll
  for (int tn = 0; tn < 8; ++tn) {
    float bi = sbg1[tn * 16 + m16];
#pragma unroll
    for (int e = 0; e < 8; ++e)
      abuf[(e + 8 * h) * 132 + tn * 16 + m16] = fmaxf(acc[tn][e] + bi, 0.f);
  }

  // ---- a = hgem @ Wg2 + bg2
  v8f acc2[8] = {};
  for (int s = 0; s < 32; ++s) {
    int k0 = 4 * s + 2 * h;
    v2f a;
    a.x = abuf[m16 * 132 + k0];
    a.y = abuf[m16 * 132 + k0 + 1];
#pragma unroll
    for (int tn = 0; tn < 8; ++tn) {
      v2f b;
      b.x = sWg2[k0 * 128 + tn * 16 + m16];
      b.y = sWg2[(k0 + 1) * 128 + tn * 16 + m16];
      acc2[tn] = __builtin_amdgcn_wmma_f32_16x16x4_f32(false, a, false, b, (short)0,
                                                       acc2[tn], false, false);
    }
  }

  // ---- softmax over the 16 neighbors (rows) per channel, then aggregate
#pragma unroll
  for (int tn = 0; tn < 8; ++tn) {
    int c = tn * 16 + m16;
    float bi = sbg2[c];
    v8f av = acc2[tn];
    float mx = -1e30f;
#pragma unroll
    for (int e = 0; e < 8; ++e) {
      av[e] += bi;
      mx = fmaxf(mx, av[e]);
    }
    mx = fmaxf(mx, __shfl_xor(mx, 16, 32));
    float sm = 0.f;
#pragma unroll
    for (int e = 0; e < 8; ++e) {
      av[e] = __expf(av[e] - mx);
      sm += av[e];
    }
    sm += __shfl_xor(sm, 16, 32);
    float rs = 1.0f / sm;
    float pa = 0.f;
#pragma unroll
    for (int e = 0; e < 8; ++e) {
      int g = sg[w * 16 + e + 8 * h];
      float vv = vmat[(size_t)g * HCH + c] + dacc[tn][e];
      pa += (av[e] * rs) * vv;
    }
    pa += __shfl_xor(pa, 16, 32);
    if (h == 0) agg[(size_t)i * HCH + c] = pa;
  }
}

// -------------------------------------------------------------------------
// 5) out = bn(u) + x
// -------------------------------------------------------------------------
__global__ void __launch_bounds__(256) k_finalize(const float* __restrict__ u,
                                                  const float* __restrict__ scale,
                                                  const float* __restrict__ shift,
                                                  const float* __restrict__ x,
                                                  float* __restrict__ out, int total,
                                                  int C) {
  int t = blockIdx.x * 256 + threadIdx.x;
  if (t < total) {
    int c = t % C;
    out[t] = u[t] * scale[c] + shift[c] + x[t];
  }
}

// -------------------------------------------------------------------------
extern "C" void kernel_launch(void* const* d_in, const int* in_sizes, int n_in,
                              void* d_out, int out_size, void* d_ws, size_t ws_size,
                              hipStream_t stream) {
  (void)in_sizes; (void)n_in; (void)out_size; (void)ws_size;
  const int* coords = (const int*)d_in[0];
  const float* x = (const float*)d_in[1];
  const float* W_top = (const float*)d_in[2];
  const float* b_top = (const float*)d_in[3];
  const float* g1 = (const float*)d_in[4];
  const float* be1 = (const float*)d_in[5];
  const float* W_phi = (const float*)d_in[6];
  const float* W_psi = (const float*)d_in[7];
  const float* W_alpha = (const float*)d_in[8];
  const float* Wd1 = (const float*)d_in[9];
  const float* bd1 = (const float*)d_in[10];
  const float* Wd2 = (const float*)d_in[11];
  const float* bd2 = (const float*)d_in[12];
  const float* Wg1 = (const float*)d_in[13];
  const float* bg1 = (const float*)d_in[14];
  const float* Wg2 = (const float*)d_in[15];
  const float* bg2 = (const float*)d_in[16];
  const float* W_down = (const float*)d_in[17];
  const float* b_down = (const float*)d_in[18];
  const float* g2 = (const float*)d_in[19];
  const float* be2 = (const float*)d_in[20];

  char* ws = (char*)d_ws;
  size_t off = 0;
  auto carve = [&](size_t bytes) -> char* {
    char* p = ws + off;
    off += (bytes + 255) & ~(size_t)255;
    return p;
  };
  int* gidx = (int*)carve((size_t)BN_ * KNN * 4);
  float* t = (float*)carve((size_t)BN_ * HCH * 4);
  float* q = (float*)carve((size_t)BN_ * HCH * 4);
  float* kf = (float*)carve((size_t)BN_ * HCH * 4);
  float* v = (float*)carve((size_t)BN_ * HCH * 4);
  float* agg = (float*)carve((size_t)BN_ * HCH * 4);
  float* u = (float*)carve((size_t)BN_ * IND * 4);
  float* scale1 = (float*)carve(HCH * 4);
  float* shift1 = (float*)carve(HCH * 4);
  float* scale2 = (float*)carve(IND * 4);
  float* shift2 = (float*)carve(IND * 4);

  // 1) neighbor indices
  k_ballquery<<<(B_ * N_) / 256, 256, 3 * N_ * 4, stream>>>(coords, gidx);

  // 2) t = x @ W_top + b_top
  k_gemm<8><<<BN_ / 128, 256, IND * HCH * 4, stream>>>(x, nullptr, nullptr, W_top,
                                                       b_top, t, BN_, IND);
  // 3) BN1 stats
  k_bnstats<<<HCH, 256, 0, stream>>>(t, g1, be1, scale1, shift1, BN_, HCH);

  // 4) q/k/v = bn(t) @ {W_phi, W_psi, W_alpha}   (BN affine fused into A)
  k_gemm<8><<<BN_ / 128, 256, HCH * HCH * 4, stream>>>(t, scale1, shift1, W_phi,
                                                       nullptr, q, BN_, HCH);
  k_gemm<8><<<BN_ / 128, 256, HCH * HCH * 4, stream>>>(t, scale1, shift1, W_psi,
                                                       nullptr, kf, BN_, HCH);
  k_gemm<8><<<BN_ / 128, 256, HCH * HCH * 4, stream>>>(t, scale1, shift1, W_alpha,
                                                       nullptr, v, BN_, HCH);

  // 5) fused attention core (one wave per point)
  size_t ptlds = (size_t)(16384 * 3 + 384 + 4 * 128 + 8 * 16 * 132) * 4 + 8 * 16 * 4;
  k_pt_core<<<BN_ / 8, 256, ptlds, stream>>>(coords, gidx, q, kf, v, Wd1, bd1, Wd2,
                                             bd2, Wg1, bg1, Wg2, bg2, agg);

  // 6) u = agg @ W_down + b_down
  k_gemm<4><<<BN_ / 128, 256, HCH * IND * 4, stream>>>(agg, nullptr, nullptr, W_down,
                                                       b_down, u, BN_, HCH);
  // 7) BN2 stats
  k_bnstats<<<IND, 256, 0, stream>>>(u, g2, be2, scale2, shift2, BN_, IND);

  // 8) out = bn(u) + x
  k_finalize<<<(BN_ * IND + 255) / 256, 256, 0, stream>>>(u, scale2, shift2, x,
                                                          (float*)d_out, BN_ * IND,
                                                          IND);
}